// DifferentiableProjectionLayer_64957085384831
// MI455X (gfx1250) — compile-verified
//
#include <hip/hip_runtime.h>
#include <hip/hip_bf16.h>
#include <math.h>

typedef __attribute__((ext_vector_type(2))) float v2f;
typedef __attribute__((ext_vector_type(8))) float v8f;

#define NQ 4096
#define NV 3200
#define NF 6240
#define QT (NQ/16)   // 256 query tiles
#define FT (NF/16)   // 390 face tiles
#define SPLIT 8      // face-range splits per query tile -> 2048 waves

// Workspace layout (floats):
//   Apanel : QT * 3 chunks * 64            = 49152
//   Bpanel : (FT+2) * 12 * 64 (2 dummy)    = 301056
//   tminbits (uint) : NQ                   = 4096
#define BPAN_OFF (QT*3*64)
#define TMIN_OFF (BPAN_OFF + (FT+2)*12*64)

// ---------------------------------------------------------------------------
// Kernel 1: per-face precompute -> B panel in WMMA B-fragment layout.
// Columns per face (det,u,v,t) over features [dirs, x×dirs, x, 1] (K=12):
//   det: rows0-2 = e2×e1
//   u  : rows0-2 = v0×e2, rows3-5 = e2
//   v  : rows0-2 = e1×v0, rows3-5 = -e1
//   t  : rows6-8 = e1×e2, row9 = -v0·(e1×e2)
// ---------------------------------------------------------------------------
__global__ void face_pre(const float* __restrict__ verts,
                         const int* __restrict__ faces,
                         float* __restrict__ Bpan) {
    int f = blockIdx.x * blockDim.x + threadIdx.x;
    if (f >= NF) return;
    int ia = faces[3*f+0], ib = faces[3*f+1], ic = faces[3*f+2];
    float p0x = verts[3*ia], p0y = verts[3*ia+1], p0z = verts[3*ia+2];
    float p1x = verts[3*ib], p1y = verts[3*ib+1], p1z = verts[3*ib+2];
    float p2x = verts[3*ic], p2y = verts[3*ic+1], p2z = verts[3*ic+2];
    float e1x = p1x-p0x, e1y = p1y-p0y, e1z = p1z-p0z;
    float e2x = p2x-p0x, e2y = p2y-p0y, e2z = p2z-p0z;
    float n3x = e2y*e1z - e2z*e1y, n3y = e2z*e1x - e2x*e1z, n3z = e2x*e1y - e2y*e1x;
    float wux = p0y*e2z - p0z*e2y, wuy = p0z*e2x - p0x*e2z, wuz = p0x*e2y - p0y*e2x;
    float wvx = e1y*p0z - e1z*p0y, wvy = e1z*p0x - e1x*p0z, wvz = e1x*p0y - e1y*p0x;
    float mx  = e1y*e2z - e1z*e2y, my  = e1z*e2x - e1x*e2z, mz  = e1x*e2y - e1y*e2x;
    float ct  = -(p0x*mx + p0y*my + p0z*mz);

    float B[4][12];
    #pragma unroll
    for (int o = 0; o < 4; ++o)
        #pragma unroll
        for (int K = 0; K < 12; ++K) B[o][K] = 0.0f;
    B[0][0]=n3x; B[0][1]=n3y; B[0][2]=n3z;
    B[1][0]=wux; B[1][1]=wuy; B[1][2]=wuz; B[1][3]=e2x;  B[1][4]=e2y;  B[1][5]=e2z;
    B[2][0]=wvx; B[2][1]=wvy; B[2][2]=wvz; B[2][3]=-e1x; B[2][4]=-e1y; B[2][5]=-e1z;
    B[3][6]=mx;  B[3][7]=my;  B[3][8]=mz;  B[3][9]=ct;

    int tb = f >> 4, n = f & 15;
    #pragma unroll
    for (int o = 0; o < 4; ++o) {
        #pragma unroll
        for (int K = 0; K < 12; ++K) {
            int kch = K >> 2, kl = K & 3, khalf = kl >> 1, p = kl & 1;
            int lane = (khalf << 4) | n;
            Bpan[(tb*12 + o*3 + kch)*64 + lane*2 + p] = B[o][K];
        }
    }
}

// ---------------------------------------------------------------------------
// Kernel 2: per-query KNN over LDS-resident vertices, blended normal,
// writes n to output, A fragments, and initializes the atomic-min buffer.
// ---------------------------------------------------------------------------
__global__ __launch_bounds__(256) void knn_normals(
        const float* __restrict__ x, const float* __restrict__ verts,
        const float* __restrict__ vnorm, float* __restrict__ out,
        float* __restrict__ Apan, unsigned int* __restrict__ tminbits) {
    __shared__ float lv[NV * 3];   // 38.4 KB of 320 KB/WGP
    for (int i = threadIdx.x; i < NV*3; i += blockDim.x) lv[i] = verts[i];
    __syncthreads();

    int q = blockIdx.x * blockDim.x + threadIdx.x;
    if (q >= NQ) return;
    tminbits[q] = 0x7F7FFFFFu;     // +FLT_MAX sentinel
    float xx = x[3*q], xy = x[3*q+1], xz = x[3*q+2];

    float dk[8]; int ik[8];
    #pragma unroll
    for (int j = 0; j < 8; ++j) { dk[j] = 3.0e38f; ik[j] = 0; }

    for (int v = 0; v < NV; ++v) {
        float dx = xx - lv[3*v], dy = xy - lv[3*v+1], dz = xz - lv[3*v+2];
        float d2 = dx*dx + dy*dy + dz*dz;
        if (d2 < dk[7]) {
            float cd = d2; int ci = v;
            #pragma unroll
            for (int j = 0; j < 8; ++j) {
                bool sw = cd < dk[j];
                float td = dk[j]; int ti = ik[j];
                dk[j] = sw ? cd : td;  ik[j] = sw ? ci : ti;
                cd    = sw ? td : cd;  ci    = sw ? ti : ci;
            }
        }
    }

    float W = 1.0f / 0.01f;
    float tnx = 0.f, tny = 0.f, tnz = 0.f;
    #pragma unroll
    for (int j = 0; j < 8; ++j) {
        float d2 = fmaxf(dk[j], 1e-8f);
        float inv = 1.0f / d2;
        W += inv;
        int id = ik[j];
        tnx += vnorm[3*id]   * inv;
        tny += vnorm[3*id+1] * inv;
        tnz += vnorm[3*id+2] * inv;
    }
    int i0 = ik[0];
    float rx = xx - lv[3*i0], ry = xy - lv[3*i0+1], rz = xz - lv[3*i0+2];
    float d2v1 = fmaxf(rx*rx + ry*ry + rz*rz, 1e-8f);
    float invdir = 1.0f / (0.01f * d2v1);
    tnx += rx * invdir; tny += ry * invdir; tnz += rz * invdir;
    float invW = 1.0f / W;
    tnx *= invW; tny *= invW; tnz *= invW;
    float nrm = sqrtf(tnx*tnx + tny*tny + tnz*tnz) + 1e-8f;
    float nx = tnx / nrm, ny = tny / nrm, nz = tnz / nrm;

    out[16384 + 3*q + 0] = nx;
    out[16384 + 3*q + 1] = ny;
    out[16384 + 3*q + 2] = nz;

    float dxr = -nx, dyr = -ny, dzr = -nz;       // dirs
    float cxx = xy*dzr - xz*dyr;                 // cx = x × dirs
    float cxy = xz*dxr - xx*dzr;
    float cxz = xx*dyr - xy*dxr;
    float F[12] = {dxr,dyr,dzr, cxx,cxy,cxz, xx,xy,xz, 1.0f, 0.0f, 0.0f};

    int qt = q >> 4, m = q & 15;
    #pragma unroll
    for (int K = 0; K < 12; ++K) {
        int kch = K >> 2, kl = K & 3, khalf = kl >> 1, p = kl & 1;
        int lane = (khalf << 4) | m;
        Apan[(qt*3 + kch)*64 + lane*2 + p] = F[K];
    }
}

// ---------------------------------------------------------------------------
// Kernel 3: one wave per (query tile, face split). 12 WMMAs per 16q x 16f
// tile; strictly branch-free epilogue (bitwise & combines, no short-circuit
// so no EXEC-mask branches between WMMA blocks); partial minima merged with
// global_atomic_min_u32 on float bit patterns.
// ---------------------------------------------------------------------------
__global__ __launch_bounds__(32) void raycast(
        const float* __restrict__ Apan, const float* __restrict__ Bpan,
        unsigned int* __restrict__ tminbits) {
    int qt = blockIdx.x / SPLIT;
    int sp = blockIdx.x % SPLIT;
    int l  = threadIdx.x;
    int tb0 = (FT * sp) / SPLIT;
    int tb1 = (FT * (sp + 1)) / SPLIT;

    v2f a0 = *(const v2f*)(Apan + (qt*3 + 0)*64 + l*2);
    v2f a1 = *(const v2f*)(Apan + (qt*3 + 1)*64 + l*2);
    v2f a2 = *(const v2f*)(Apan + (qt*3 + 2)*64 + l*2);

    float bn[8], bd[8];          // running min as quotient pair bn/bd
    #pragma unroll
    for (int r = 0; r < 8; ++r) { bn[r] = 3.0e38f; bd[r] = 1.0f; }

    v2f b[12];
    #pragma unroll
    for (int i = 0; i < 12; ++i)
        b[i] = *(const v2f*)(Bpan + (tb0*12 + i)*64 + l*2);

    for (int tb = tb0; tb < tb1; ++tb) {
        v8f dacc[4];
        #pragma unroll
        for (int o = 0; o < 4; ++o) {
            v8f c = {0,0,0,0,0,0,0,0};
            c = __builtin_amdgcn_wmma_f32_16x16x4_f32(false, a0, false, b[o*3+0], (short)0, c, false, false);
            c = __builtin_amdgcn_wmma_f32_16x16x4_f32(false, a1, false, b[o*3+1], (short)0, c, false, false);
            c = __builtin_amdgcn_wmma_f32_16x16x4_f32(false, a2, false, b[o*3+2], (short)0, c, false, false);
            dacc[o] = c;
        }
        // prefetch next tile (distance-1 double buffer; tile FT is a dummy)
        v2f b2[12];
        #pragma unroll
        for (int i = 0; i < 12; ++i)
            b2[i] = *(const v2f*)(Bpan + ((tb+1)*12 + i)*64 + l*2);

        #pragma unroll
        for (int r = 0; r < 8; ++r) {
            float det = dacc[0][r];
            float ad  = fabsf(det);
            float sg  = (det < 0.0f) ? -1.0f : 1.0f;
            float us  = dacc[1][r] * sg;
            float vs  = dacc[2][r] * sg;
            float ts  = dacc[3][r] * sg;
            const float tol = 1e-6f;
            float ntol = -tol * ad;
            // bitwise & (no short-circuit) -> flat v_cmp/s_and, no branches
            int valid = (int)(ad > 1e-9f) & (int)(us >= ntol) & (int)(vs >= ntol)
                      & (int)(us + vs <= (1.0f + tol) * ad) & (int)(ts > tol * ad);
            // ts/ad < bn/bd  <=>  ts*bd < bn*ad  (all positive)
            int less = valid & (int)(ts * bd[r] < bn[r] * ad);
            bn[r] = less ? ts : bn[r];
            bd[r] = less ? ad : bd[r];
        }
        #pragma unroll
        for (int i = 0; i < 12; ++i) b[i] = b2[i];
    }

    int half = l >> 4, lane15 = l & 15;
    #pragma unroll
    for (int r = 0; r < 8; ++r) {
        float t = bn[r] / bd[r];                 // one divide per query
        t = fminf(t, __shfl_xor(t, 1, 32));
        t = fminf(t, __shfl_xor(t, 2, 32));
        t = fminf(t, __shfl_xor(t, 4, 32));
        t = fminf(t, __shfl_xor(t, 8, 32));
        if (lane15 == r) {
            int q = qt*16 + r + 8*half;
            atomicMin(&tminbits[q], __float_as_uint(t));
        }
    }
}

// ---------------------------------------------------------------------------
// Kernel 4: finalize xc and s from the merged t_min.
// ---------------------------------------------------------------------------
__global__ void finalize(const float* __restrict__ x,
                         const unsigned int* __restrict__ tminbits,
                         float* __restrict__ out) {
    int q = blockIdx.x * blockDim.x + threadIdx.x;
    if (q >= NQ) return;
    float tm = __uint_as_float(tminbits[q]);
    if (tm > 1.0e37f) tm = 0.0f;                 // no finite hit -> 0
    float xx = x[3*q], xy = x[3*q+1], xz = x[3*q+2];
    float nx = out[16384+3*q], ny = out[16384+3*q+1], nz = out[16384+3*q+2];
    float xcx = xx - tm*nx, xcy = xy - tm*ny, xcz = xz - tm*nz;   // dirs = -n
    out[3*q+0] = xcx; out[3*q+1] = xcy; out[3*q+2] = xcz;
    out[12288 + q] = (xx-xcx)*nx + (xy-xcy)*ny + (xz-xcz)*nz;
}

extern "C" void kernel_launch(void* const* d_in, const int* in_sizes, int n_in,
                              void* d_out, int out_size, void* d_ws, size_t ws_size,
                              hipStream_t stream) {
    const float* x     = (const float*)d_in[0];
    const float* verts = (const float*)d_in[1];
    const float* vnorm = (const float*)d_in[2];
    const int*   faces = (const int*)d_in[3];
    float* out  = (float*)d_out;
    float* Apan = (float*)d_ws;
    float* Bpan = Apan + BPAN_OFF;
    unsigned int* tminbits = (unsigned int*)(Apan + TMIN_OFF);

    face_pre<<<(NF + 255)/256, 256, 0, stream>>>(verts, faces, Bpan);
    knn_normals<<<NQ/256, 256, 0, stream>>>(x, verts, vnorm, out, Apan, tminbits);
    raycast<<<QT * SPLIT, 32, 0, stream>>>(Apan, Bpan, tminbits);
    finalize<<<NQ/256, 256, 0, stream>>>(x, tminbits, out);
}